// PPOActorCritic_GNN_new_68358699483727
// MI455X (gfx1250) — compile-verified
//
#include <hip/hip_runtime.h>
#include <hip/hip_bf16.h>
#include <math.h>

typedef __attribute__((ext_vector_type(2))) float v2f;
typedef __attribute__((ext_vector_type(8))) float v8f;

// fast sigmoid: v_exp_f32 + v_rcp_f32 (avoids IEEE div_scale/div_fixup chain)
__device__ __forceinline__ float sigmoidf(float x) {
  return __builtin_amdgcn_rcpf(1.0f + __expf(-x));
}

__device__ __forceinline__ float waveReduceMax(float v) {
  #pragma unroll
  for (int o = 16; o > 0; o >>= 1) v = fmaxf(v, __shfl_xor(v, o, 32));
  return v;
}
__device__ __forceinline__ float waveReduceSum(float v) {
  #pragma unroll
  for (int o = 16; o > 0; o >>= 1) v += __shfl_xor(v, o, 32);
  return v;
}

// ordered-int encoding for float atomic max over signed floats
__device__ __forceinline__ unsigned fkey(float f) {
  unsigned u = __float_as_uint(f);
  return (u & 0x80000000u) ? ~u : (u | 0x80000000u);
}
__device__ __forceinline__ float fkey_dec(unsigned k) {
  unsigned u = (k & 0x80000000u) ? (k & 0x7FFFFFFFu) : ~k;
  return __uint_as_float(u);
}

// ---------------- K1: zero accumulators + scalars ----------------
__global__ void k_zero(float* __restrict__ accum, float* __restrict__ scalars, int n8) {
  int i = blockIdx.x * blockDim.x + threadIdx.x;
  if (i < n8) accum[i] = 0.0f;
  if (i < 16) scalars[i] = 0.0f;
}

// ---------------- K2: node features + column-max (abs) ----------------
__global__ void k_features(const float* __restrict__ hop, const float* __restrict__ rd,
                           const float* __restrict__ ett, const float* __restrict__ ct,
                           const float* __restrict__ c2n, const float* __restrict__ c2t,
                           float* __restrict__ feat, unsigned* __restrict__ colmax, int n) {
  int i = blockIdx.x * blockDim.x + threadIdx.x;
  float f1 = 0.f, f2 = 0.f, f3 = 0.f, f4 = 0.f;
  if (i < n) {
    float ax = c2n[3*i+0], ay = c2n[3*i+1], az = c2n[3*i+2];
    float bx = c2t[3*i+0], by = c2t[3*i+1], bz = c2t[3*i+2];
    float dot = ax*bx + ay*by + az*bz;
    float na = fmaxf(sqrtf(ax*ax + ay*ay + az*az), 1e-6f);
    float nb = fmaxf(sqrtf(bx*bx + by*by + bz*bz), 1e-6f);
    float cs = dot / (na * nb);
    f1 = hop[i]; f2 = cs; f3 = rd[i]; f4 = ett[i];
    size_t b = (size_t)i * 8;
    feat[b+0] = ct[i] * (1.0f/1440.0f);
    feat[b+1] = f1; feat[b+2] = f2; feat[b+3] = f3; feat[b+4] = f4;
  }
  float m1 = waveReduceMax(fabsf(f1));
  float m2 = waveReduceMax(fabsf(f2));
  float m3 = waveReduceMax(fabsf(f3));
  float m4 = waveReduceMax(fabsf(f4));
  if ((threadIdx.x & 31) == 0) {
    atomicMax(&colmax[0], __float_as_uint(m1));  // values are >=0: uint order == float order
    atomicMax(&colmax[1], __float_as_uint(m2));
    atomicMax(&colmax[2], __float_as_uint(m3));
    atomicMax(&colmax[3], __float_as_uint(m4));
  }
}

// ---------------- K3: inf-normalize columns 1..4 ----------------
__global__ void k_normalize(float* __restrict__ feat, const unsigned* __restrict__ colmax, int n) {
  int i = blockIdx.x * blockDim.x + threadIdx.x;
  if (i >= n) return;
  size_t b = (size_t)i * 8;
  #pragma unroll
  for (int c = 0; c < 4; ++c) {
    float m = fmaxf(__uint_as_float(colmax[c]), 1e-12f);
    feat[b + 1 + c] *= (1.0f / m);
  }
}

// ---------------- K4: edge mean-aggregation (scatter-add) ----------------
__global__ void k_edges(const int* __restrict__ src, const int* __restrict__ dst,
                        const float* __restrict__ feat, float* __restrict__ accum, int e) {
  int i = blockIdx.x * blockDim.x + threadIdx.x;
  if (i >= e) return;
  int u = src[i], v = dst[i];
  const float* fu = feat + (size_t)u * 8;
  float* av = accum + (size_t)v * 8;
  atomicAdd(av + 0, fu[0]);
  atomicAdd(av + 1, fu[1]);
  atomicAdd(av + 2, fu[2]);
  atomicAdd(av + 3, fu[3]);
  atomicAdd(av + 4, fu[4]);
  atomicAdd(av + 5, 1.0f);
}

// ---------------- K5: concat=(feat+x)/2 in place; critic node sums ----------------
__global__ void k_concat(const float* __restrict__ feat, float* __restrict__ accum,
                         float* __restrict__ scalars, int n) {
  int i = blockIdx.x * blockDim.x + threadIdx.x;
  float f[5] = {0,0,0,0,0}, x[5] = {0,0,0,0,0};
  if (i < n) {
    size_t b = (size_t)i * 8;
    float cnt = accum[b + 5];
    float inv = (cnt > 0.0f) ? (1.0f / cnt) : 0.0f;
    #pragma unroll
    for (int c = 0; c < 5; ++c) {
      f[c] = feat[b + c];
      x[c] = accum[b + c] * inv;
      accum[b + c] = 0.5f * (f[c] + x[c]);   // becomes padded A-matrix row
    }
    accum[b + 5] = 0.0f; accum[b + 6] = 0.0f; accum[b + 7] = 0.0f;  // K padding 5->8
  }
  // critic features are feat/x columns {0,1,3}
  float s0 = waveReduceSum(f[0]);
  float s1 = waveReduceSum(f[1]);
  float s2 = waveReduceSum(f[3]);
  float s3 = waveReduceSum(x[0]);
  float s4 = waveReduceSum(x[1]);
  float s5 = waveReduceSum(x[3]);
  if ((threadIdx.x & 31) == 0) {
    atomicAdd(&scalars[4], s0); atomicAdd(&scalars[5], s1); atomicAdd(&scalars[6], s2);
    atomicAdd(&scalars[7], s3); atomicAdd(&scalars[8], s4); atomicAdd(&scalars[9], s5);
  }
}

// ---------------- K6: actor MLP with f32 WMMA, one wave = 16 nodes ----------------
__global__ __launch_bounds__(128) void k_mlp(const float* __restrict__ concat,
                                             const float* __restrict__ aW2, const float* __restrict__ ab2,
                                             const float* __restrict__ aW3, const float* __restrict__ ab3,
                                             const float* __restrict__ aW4, const float* __restrict__ ab4,
                                             float* __restrict__ logits, int n) {
  __shared__ float lds[4 * 3072];                 // per-wave: h1[16*128] + h2[16*64]
  const int tid  = threadIdx.x;
  const int wave = tid >> 5;
  const int lane = tid & 31;
  const int half = lane >> 4;                     // selects K pair (A) / K pair (B) / M+8 (C,D)
  const int l16  = lane & 15;
  const long gwave = (long)blockIdx.x * 4 + wave;
  const long m0 = gwave * 16;
  if (m0 >= n) return;                            // wave-uniform: EXEC stays all-ones for WMMA

  float* h1 = &lds[wave * 3072];
  float* h2 = h1 + 2048;

  // ---- A fragments: rows m0..m0+15, K padded to 8 (cols 5..7 are zero) ----
  long mrow = m0 + l16; if (mrow > (long)n - 1) mrow = (long)n - 1;
  const float* Crow = concat + (size_t)mrow * 8;
  v2f a0, a1;
  a0.x = Crow[half * 2 + 0];  a0.y = Crow[half * 2 + 1];        // K = 0..3
  a1.x = Crow[4 + half * 2];  a1.y = Crow[5 + half * 2];        // K = 4..7

  // ---- layer 1: [16,8(5)] @ aW2[8(5),128] -> sigmoid -> h1 in LDS ----
  for (int t = 0; t < 8; ++t) {
    int col = t * 16 + l16;
    v2f b0, b1;
    b0.x = aW2[(half * 2 + 0) * 128 + col];
    b0.y = aW2[(half * 2 + 1) * 128 + col];
    b1.x = half ? 0.0f : aW2[4 * 128 + col];                    // row 4 real, rows 5..7 zero
    b1.y = 0.0f;
    v8f acc = {};
    acc = __builtin_amdgcn_wmma_f32_16x16x4_f32(false, a0, false, b0, (short)0, acc, false, false);
    acc = __builtin_amdgcn_wmma_f32_16x16x4_f32(false, a1, false, b1, (short)0, acc, false, false);
    float bias = ab2[col];
    #pragma unroll
    for (int r = 0; r < 8; ++r)
      h1[(r + half * 8) * 128 + col] = sigmoidf(acc[r] + bias);
  }

  // ---- layer 2: [16,128] @ aW3[128,64] -> sigmoid -> h2 in LDS ----
  const float* h1row = h1 + l16 * 128;
  for (int t = 0; t < 4; ++t) {
    int col = t * 16 + l16;
    v8f acc = {};
    #pragma unroll 4
    for (int k4 = 0; k4 < 32; ++k4) {
      int kb = k4 * 4 + half * 2;
      v2f a, b;
      a.x = h1row[kb];
      a.y = h1row[kb + 1];
      b.x = aW3[(kb + 0) * 64 + col];
      b.y = aW3[(kb + 1) * 64 + col];
      acc = __builtin_amdgcn_wmma_f32_16x16x4_f32(false, a, false, b, (short)0, acc, false, false);
    }
    float bias = ab3[col];
    #pragma unroll
    for (int r = 0; r < 8; ++r)
      h2[(r + half * 8) * 64 + col] = sigmoidf(acc[r] + bias);
  }

  // ---- layer 3: [16,64] @ aW4[64,1] — all 32 lanes: each half does 32 cols ----
  {
    float s = 0.0f;
    const float* hrow = h2 + l16 * 64 + half * 32;
    const float* w    = aW4 + half * 32;
    #pragma unroll 8
    for (int c = 0; c < 32; ++c) s += hrow[c] * w[c];
    s += __shfl_xor(s, 16, 32);                  // combine the two half-row partials
    if (lane < 16) {
      long m = m0 + l16;
      if (m < n) logits[m] = s + ab4[0];
    }
  }
}

// ---------------- K7/K8/K9: masked global softmax ----------------
__global__ void k_smax_max(const float* __restrict__ logits, const int* __restrict__ currp,
                           unsigned* __restrict__ maxkey, int n) {
  int i = blockIdx.x * blockDim.x + threadIdx.x;
  int curr = currp[0];
  float v = -3.4e38f;
  if (i < n && i != curr) v = logits[i];
  v = waveReduceMax(v);
  if ((threadIdx.x & 31) == 0) atomicMax(maxkey, fkey(v));
}

__global__ void k_smax_sum(const float* __restrict__ logits, const int* __restrict__ currp,
                           const unsigned* __restrict__ maxkey, float* __restrict__ sump, int n) {
  int i = blockIdx.x * blockDim.x + threadIdx.x;
  int curr = currp[0];
  float mx = fkey_dec(*maxkey);
  float e = 0.0f;
  if (i < n && i != curr) e = __expf(logits[i] - mx);
  e = waveReduceSum(e);
  if ((threadIdx.x & 31) == 0) atomicAdd(sump, e);
}

__global__ void k_smax_norm(const float* __restrict__ logits, const int* __restrict__ currp,
                            const unsigned* __restrict__ maxkey, const float* __restrict__ sump,
                            float* __restrict__ out, int n) {
  int i = blockIdx.x * blockDim.x + threadIdx.x;
  if (i >= n) return;
  int curr = currp[0];
  float mx = fkey_dec(*maxkey);
  float inv = 1.0f / sump[0];
  out[i] = (i == curr) ? 0.0f : __expf(logits[i] - mx) * inv;
}

// ---------------- K10: critic MLP (single wave) ----------------
__global__ void k_critic(const float* __restrict__ scalars,
                         const float* __restrict__ cW2, const float* __restrict__ cb2,
                         const float* __restrict__ cW3, const float* __restrict__ cb3,
                         const float* __restrict__ cW4, const float* __restrict__ cb4,
                         float* __restrict__ out, int n) {
  __shared__ float h1[128];
  __shared__ float h2[64];
  int lane = threadIdx.x;                 // blockDim.x == 32
  float invN = 1.0f / (float)n;
  float cx[6];
  #pragma unroll
  for (int i = 0; i < 6; ++i) cx[i] = scalars[4 + i] * invN;
  for (int j = lane; j < 128; j += 32) {
    float s = cb2[j];
    #pragma unroll
    for (int i = 0; i < 6; ++i) s += cx[i] * cW2[i * 128 + j];
    h1[j] = sigmoidf(s);
  }
  __syncthreads();
  for (int j = lane; j < 64; j += 32) {
    float s = cb3[j];
    for (int i = 0; i < 128; ++i) s += h1[i] * cW3[i * 64 + j];
    h2[j] = sigmoidf(s);
  }
  __syncthreads();
  float s = h2[lane] * cW4[lane] + h2[lane + 32] * cW4[lane + 32];
  s = waveReduceSum(s);
  if (lane == 0) out[n] = s + cb4[0];
}

extern "C" void kernel_launch(void* const* d_in, const int* in_sizes, int n_in,
                              void* d_out, int out_size, void* d_ws, size_t ws_size,
                              hipStream_t stream) {
  const int n = in_sizes[0];
  const int e = in_sizes[7];

  const float* hop = (const float*)d_in[0];
  const float* rd  = (const float*)d_in[1];
  const float* ett = (const float*)d_in[2];
  const float* ct  = (const float*)d_in[3];
  const float* c2n = (const float*)d_in[4];
  const float* c2t = (const float*)d_in[5];
  const int*   src = (const int*)d_in[7];
  const int*   dst = (const int*)d_in[8];
  const int*   cur = (const int*)d_in[9];
  const float* aW2 = (const float*)d_in[10];
  const float* ab2 = (const float*)d_in[11];
  const float* aW3 = (const float*)d_in[12];
  const float* ab3 = (const float*)d_in[13];
  const float* aW4 = (const float*)d_in[14];
  const float* ab4 = (const float*)d_in[15];
  const float* cW2 = (const float*)d_in[16];
  const float* cb2 = (const float*)d_in[17];
  const float* cW3 = (const float*)d_in[18];
  const float* cb3 = (const float*)d_in[19];
  const float* cW4 = (const float*)d_in[20];
  const float* cb4 = (const float*)d_in[21];

  float* out = (float*)d_out;
  float* ws  = (float*)d_ws;

  // Workspace layout (floats):
  //   [0,            n*8)   feat [N][8]  (cols 0..4 used); later reused as logits[N]
  //   [n*8,          n*16)  accum [N][8] (sums 0..4, count 5) -> rewritten to padded concat
  //   [n*16,         n*16+16) scalars: [0..3] colmax bits, [4..9] critic sums,
  //                           [10] softmax max key (uint), [11] softmax sum
  float*    feat    = ws;
  float*    accum   = ws + (size_t)n * 8;
  float*    scalars = ws + (size_t)n * 16;
  unsigned* colmax  = (unsigned*)scalars;
  unsigned* maxkey  = (unsigned*)(scalars + 10);
  float*    sump    = scalars + 11;
  float*    logits  = ws;  // reuse feat region after it is consumed

  const int B = 256;
  const int n8 = n * 8;

  k_zero     <<<(n8 + B - 1) / B, B, 0, stream>>>(accum, scalars, n8);
  k_features <<<(n + B - 1) / B, B, 0, stream>>>(hop, rd, ett, ct, c2n, c2t, feat, colmax, n);
  k_normalize<<<(n + B - 1) / B, B, 0, stream>>>(feat, colmax, n);
  k_edges    <<<(e + B - 1) / B, B, 0, stream>>>(src, dst, feat, accum, e);
  k_concat   <<<(n + B - 1) / B, B, 0, stream>>>(feat, accum, scalars, n);

  const int waves  = (n + 15) / 16;
  const int blocks = (waves + 3) / 4;
  k_mlp      <<<blocks, 128, 0, stream>>>(accum, aW2, ab2, aW3, ab3, aW4, ab4, logits, n);

  k_smax_max <<<(n + B - 1) / B, B, 0, stream>>>(logits, cur, maxkey, n);
  k_smax_sum <<<(n + B - 1) / B, B, 0, stream>>>(logits, cur, maxkey, sump, n);
  k_smax_norm<<<(n + B - 1) / B, B, 0, stream>>>(logits, cur, maxkey, sump, out, n);

  k_critic   <<<1, 32, 0, stream>>>(scalars, cW2, cb2, cW3, cb3, cW4, cb4, out, n);
}